// qnn_graph_36661840839088
// MI455X (gfx1250) — compile-verified
//
#include <hip/hip_runtime.h>

typedef __attribute__((ext_vector_type(2))) float v2f;
typedef __attribute__((ext_vector_type(8))) float v8f;

namespace {
constexpr int T = 8192, H = 32, N = 64;
constexpr int L = 32;          // chunk length (numerically safe for 1/A in fp32)
constexpr int C = T / L;       // 256 chunks
constexpr int NP = N + 4;      // LDS padding: 272B row stride = 16B aligned, conflict-free
constexpr int LP = L + 1;
}

__device__ __forceinline__ float clamp_w(float w) { return fmaxf(w, 1e-6f); }

// ---------------------------------------------------------------------------
// Kernel 1: per-(chunk, head) summary  U_c = (k ⊙ A_L/A_t)^T V   (N x N)
// One wave per block; 128 x v_wmma_f32_16x16x4_f32 per block.
// ---------------------------------------------------------------------------
__global__ __launch_bounds__(32) void wkv_chunk_summary(
    const float* __restrict__ k, const float* __restrict__ v,
    const float* __restrict__ td, float* __restrict__ U,
    float* __restrict__ wtot)
{
  const int c = blockIdx.x, h = blockIdx.y, lane = threadIdx.x;
  __shared__ float sK[L][NP];   // raw k -> k scaled by remaining-decay
  __shared__ float sV[L][NP];
  __shared__ float sA[L][NP];   // raw td -> inclusive decay prefix

  const size_t base = (((size_t)c * L) * H + h) * (size_t)N;
  const size_t step = (size_t)H * N;

  // cooperative b128 staging: 3 x 2048 floats
#pragma unroll
  for (int q = 0; q < (L * N) / (32 * 4); ++q) {
    const int e = (lane + q * 32) * 4;
    const int t = e >> 6, i = e & 63;
    const size_t g = base + (size_t)t * step + i;
    *(float4*)&sK[t][i] = *(const float4*)&k[g];
    *(float4*)&sA[t][i] = *(const float4*)&td[g];
    *(float4*)&sV[t][i] = *(const float4*)&v[g];
  }
  __syncthreads();

  // each lane owns channels i = lane, lane+32: prefix products in place
  for (int ii = 0; ii < 2; ++ii) {
    const int i = lane + ii * 32;
    float a = 1.0f;
    for (int t = 0; t < L; ++t) {
      a *= clamp_w(sA[t][i]);
      sA[t][i] = a;
    }
    wtot[((size_t)c * H + h) * N + i] = a;
    const float atot = a;
    for (int t = 0; t < L; ++t)
      sK[t][i] *= atot / sA[t][i];
  }
  __syncthreads();

  const int m = lane & 15;             // M / N fragment index
  const int kh = (lane >> 4) * 2;      // K sub-index per half-wave
  const int roff = (lane < 16) ? 0 : 8;
  const size_t ubase = (((size_t)c * H + h) * N) * (size_t)N;

  for (int ib = 0; ib < 4; ++ib) {
    for (int jb = 0; jb < 4; ++jb) {
      v8f acc = {};
#pragma unroll
      for (int kb = 0; kb < L / 4; ++kb) {
        const int kk = kb * 4 + kh;
        v2f a, b;
        a.x = sK[kk][ib * 16 + m];       // A[M=i][K=t] = K~[t][i]
        a.y = sK[kk + 1][ib * 16 + m];
        b.x = sV[kk][jb * 16 + m];       // B[K=t][N=j] = V[t][j]
        b.y = sV[kk + 1][jb * 16 + m];
        acc = __builtin_amdgcn_wmma_f32_16x16x4_f32(false, a, false, b,
                                                    (short)0, acc, false, false);
      }
#pragma unroll
      for (int p = 0; p < 8; ++p)
        U[ubase + (size_t)(ib * 16 + p + roff) * N + (jb * 16 + m)] = acc[p];
    }
  }
}

// ---------------------------------------------------------------------------
// Kernel 2: sequential inter-chunk scan (only 256 serial steps), all b128.
// In place: slot c is read (U_c) then overwritten with S_c (state at chunk
// start). Also emits the final state to d_out tail.
// ---------------------------------------------------------------------------
__global__ __launch_bounds__(256) void wkv_state_scan(
    const float* __restrict__ state0, float* __restrict__ S,
    const float* __restrict__ wtot, float* __restrict__ finalS)
{
  const int h = blockIdx.x, tid = threadIdx.x;
  float4 s[4];
#pragma unroll
  for (int g = 0; g < 4; ++g)
    s[g] = *(const float4*)&state0[(size_t)h * N * N + (g * 256 + tid) * 4];

  for (int c = 0; c < C; ++c) {
    const size_t ub = (((size_t)c * H + h) * N) * (size_t)N;
    const size_t wb = ((size_t)c * H + h) * N;
#pragma unroll
    for (int g = 0; g < 4; ++g) {
      const int e = (g * 256 + tid) * 4;      // 4 consecutive j in one row i
      const float4 u = *(const float4*)&S[ub + e];
      const float w = wtot[wb + (e >> 6)];
      *(float4*)&S[ub + e] = s[g];            // publish state at start of chunk c
      s[g].x = fmaf(w, s[g].x, u.x);
      s[g].y = fmaf(w, s[g].y, u.y);
      s[g].z = fmaf(w, s[g].z, u.z);
      s[g].w = fmaf(w, s[g].w, u.w);
    }
  }
#pragma unroll
  for (int g = 0; g < 4; ++g)
    *(float4*)&finalS[(size_t)h * N * N + (g * 256 + tid) * 4] = s[g];
}

// ---------------------------------------------------------------------------
// Kernel 3: per-(chunk, head) outputs.
// out = P'·V + (r ⊙ A_{t-1})·S_c, with P' = strict-tril((r⊙A_{t-1})(k/A_t)^T)
// plus diagonal c_t = Σ_i r tf k. One wave; 256 WMMAs per block.
// ---------------------------------------------------------------------------
__global__ __launch_bounds__(32) void wkv_chunk_output(
    const float* __restrict__ k, const float* __restrict__ v,
    const float* __restrict__ r, const float* __restrict__ td,
    const float* __restrict__ tf, const float* __restrict__ Sbuf,
    float* __restrict__ out)
{
  const int c = blockIdx.x, h = blockIdx.y, lane = threadIdx.x;
  __shared__ float sR[L][NP];    // raw r -> r * A_excl
  __shared__ float sKh[L][NP];   // raw k -> k / A_incl
  __shared__ float sV[L][NP];
  __shared__ float sTd[L][NP];   // raw td
  __shared__ float sS[N][NP];    // chunk-start state
  __shared__ float sP[L][LP];    // masked score matrix
  __shared__ float sC[L];        // diagonal terms

  const size_t base = (((size_t)c * L) * H + h) * (size_t)N;
  const size_t step = (size_t)H * N;
  const size_t sbase = (((size_t)c * H + h) * N) * (size_t)N;

  sC[lane] = 0.0f;

  // cooperative b128 staging
#pragma unroll
  for (int q = 0; q < (L * N) / (32 * 4); ++q) {
    const int e = (lane + q * 32) * 4;
    const int t = e >> 6, i = e & 63;
    const size_t g = base + (size_t)t * step + i;
    *(float4*)&sR[t][i]  = *(const float4*)&r[g];
    *(float4*)&sKh[t][i] = *(const float4*)&k[g];
    *(float4*)&sV[t][i]  = *(const float4*)&v[g];
    *(float4*)&sTd[t][i] = *(const float4*)&td[g];
  }
#pragma unroll
  for (int q = 0; q < (N * N) / (32 * 4); ++q) {
    const int e = (lane + q * 32) * 4;
    *(float4*)&sS[e >> 6][e & 63] = *(const float4*)&Sbuf[sbase + e];
  }
  __syncthreads();

  // per-channel decay pass, in place (lane owns columns i = lane, lane+32)
  for (int ii = 0; ii < 2; ++ii) {
    const int i = lane + ii * 32;
    const float tfi = tf[(size_t)h * N + i];
    float a = 1.0f;
    for (int t = 0; t < L; ++t) {
      const float w = clamp_w(sTd[t][i]);
      const float aex = a;
      a *= w;
      const float rv = sR[t][i];
      const float kv = sKh[t][i];
      sR[t][i] = rv * aex;
      sKh[t][i] = kv / a;
      atomicAdd(&sC[t], rv * tfi * kv);   // ds_add_f32
    }
  }
  __syncthreads();

  const int m = lane & 15;
  const int kh = (lane >> 4) * 2;
  const int roff = (lane < 16) ? 0 : 8;

  // scores: R~ (LxN) x K̂^T (NxL), then mask + diagonal, stage in LDS
  for (int tb = 0; tb < 2; ++tb) {
    for (int sb = 0; sb < 2; ++sb) {
      v8f acc = {};
#pragma unroll
      for (int kb = 0; kb < N / 4; ++kb) {
        const int kk = kb * 4 + kh;
        v2f a, b;
        a.x = sR[tb * 16 + m][kk];        // A[M=t][K=i]
        a.y = sR[tb * 16 + m][kk + 1];
        b.x = sKh[sb * 16 + m][kk];       // B[K=i][N=s] = K̂[s][i]
        b.y = sKh[sb * 16 + m][kk + 1];
        acc = __builtin_amdgcn_wmma_f32_16x16x4_f32(false, a, false, b,
                                                    (short)0, acc, false, false);
      }
#pragma unroll
      for (int p = 0; p < 8; ++p) {
        const int t = tb * 16 + p + roff;
        const int s = sb * 16 + m;
        sP[t][s] = (s < t) ? acc[p] : ((s == t) ? sC[t] : 0.0f);
      }
    }
  }
  __syncthreads();

  // out = P'(LxL) x V(LxN)  +  R~(LxN) x S(NxN)
  for (int tb = 0; tb < 2; ++tb) {
    for (int jb = 0; jb < 4; ++jb) {
      v8f acc = {};
#pragma unroll
      for (int kb = 0; kb < L / 4; ++kb) {
        const int kk = kb * 4 + kh;
        v2f a, b;
        a.x = sP[tb * 16 + m][kk];
        a.y = sP[tb * 16 + m][kk + 1];
        b.x = sV[kk][jb * 16 + m];
        b.y = sV[kk + 1][jb * 16 + m];
        acc = __builtin_amdgcn_wmma_f32_16x16x4_f32(false, a, false, b,
                                                    (short)0, acc, false, false);
      }
#pragma unroll
      for (int kb = 0; kb < N / 4; ++kb) {
        const int kk = kb * 4 + kh;
        v2f a, b;
        a.x = sR[tb * 16 + m][kk];
        a.y = sR[tb * 16 + m][kk + 1];
        b.x = sS[kk][jb * 16 + m];
        b.y = sS[kk + 1][jb * 16 + m];
        acc = __builtin_amdgcn_wmma_f32_16x16x4_f32(false, a, false, b,
                                                    (short)0, acc, false, false);
      }
#pragma unroll
      for (int p = 0; p < 8; ++p) {
        const int tg = c * L + tb * 16 + p + roff;
        out[((size_t)tg * H + h) * N + (jb * 16 + m)] = acc[p];
      }
    }
  }
}

// ---------------------------------------------------------------------------
extern "C" void kernel_launch(void* const* d_in, const int* in_sizes, int n_in,
                              void* d_out, int out_size, void* d_ws, size_t ws_size,
                              hipStream_t stream) {
  (void)in_sizes; (void)n_in; (void)out_size; (void)ws_size;
  const float* k  = (const float*)d_in[0];
  const float* v  = (const float*)d_in[1];
  const float* r  = (const float*)d_in[2];
  const float* st = (const float*)d_in[3];
  const float* tf = (const float*)d_in[4];
  const float* td = (const float*)d_in[5];
  float* out    = (float*)d_out;
  float* finalS = out + (size_t)T * H * N;           // outputs concatenated
  float* U      = (float*)d_ws;                      // C*H*N*N floats (-> S_c)
  float* wtot   = U + (size_t)C * H * N * N;         // C*H*N floats

  dim3 grid(C, H);
  wkv_chunk_summary<<<grid, 32, 0, stream>>>(k, v, td, U, wtot);
  wkv_state_scan<<<H, 256, 0, stream>>>(st, U, wtot, finalS);
  wkv_chunk_output<<<grid, 32, 0, stream>>>(k, v, r, td, tf, U, out);
}